// FraudDetectionModel_65481071403615
// MI455X (gfx1250) — compile-verified
//
#include <hip/hip_runtime.h>

#define NLAYER 32
#define BLOCK  256
#define PPT    4                         // points per thread
#define TPB    4                         // tiles per persistent block
#define TILE_PTS    (BLOCK * PPT)        // 1024 points per tile
#define TILE_FLOATS (TILE_PTS * 2)       // 2048 floats = 8 KB per tile
#define GRID   2048                      // 2048 blocks * 4 tiles = 8192 tiles = 2^23 points

#if __has_builtin(__builtin_amdgcn_global_load_async_to_lds_b128) && \
    __has_builtin(__builtin_amdgcn_s_wait_asynccnt)
#define USE_ASYNC 1
#else
#define USE_ASYNC 0
#endif

typedef int v4i_t __attribute__((ext_vector_type(4)));
typedef __attribute__((address_space(1))) char  gas_char;   // global, byte-addressed
typedef __attribute__((address_space(3))) char  las_char;   // LDS, byte-addressed
typedef __attribute__((address_space(1))) v4i_t gas_v4i;    // global int4
typedef __attribute__((address_space(3))) v4i_t las_v4i;    // LDS int4

// gfx1250 hardware tanh transcendental.
__device__ __forceinline__ float htanh(float x) {
#if __has_builtin(__builtin_amdgcn_tanhf)
    return __builtin_amdgcn_tanhf(x);
#elif __has_builtin(__builtin_amdgcn_tanh_f32)
    return __builtin_amdgcn_tanh_f32(x);
#else
    float r;
    asm("v_tanh_f32 %0, %1" : "=v"(r) : "v"(x));
    return r;
#endif
}

__global__ __launch_bounds__(BLOCK)
void fraud_chain_kernel(const float* __restrict__ x,     // [B,2]
                        const float* __restrict__ Ws,    // [L,2,2]
                        const float* __restrict__ bs,    // [L,2]
                        const float* __restrict__ sc,    // [L,2] scales
                        const float* __restrict__ sh,    // [L,2] shifts
                        const float* __restrict__ wo,    // [1,2]
                        const float* __restrict__ bo,    // [1]
                        float*       __restrict__ out)   // [B,1]
{
    const int tid   = threadIdx.x;
    const int tile0 = blockIdx.x * TPB;

#if USE_ASYNC
    __shared__ float lbuf[2][TILE_FLOATS];   // 2 x 8 KB double buffer

    // Async DMA one 8 KB x-tile into an LDS buffer. Each thread issues two
    // b128 copies (16 B each): 2 async instructions per wave per tile.
    auto prefetch = [&](int tile, int bufsel) {
        gas_char* g = (gas_char*)x + (size_t)tile * (TILE_FLOATS * 4);
        las_char* l = (las_char*)&lbuf[bufsel][0];
        const int o0 = tid * 16;
        const int o1 = BLOCK * 16 + tid * 16;
        __builtin_amdgcn_global_load_async_to_lds_b128((gas_v4i*)(g + o0),
                                                       (las_v4i*)(l + o0), 0, 0);
        __builtin_amdgcn_global_load_async_to_lds_b128((gas_v4i*)(g + o1),
                                                       (las_v4i*)(l + o1), 0, 0);
    };

    prefetch(tile0, 0);
#endif

    for (int k = 0; k < TPB; ++k) {
        const int tile = tile0 + k;

        float4 a, b;
#if USE_ASYNC
        // Prefetch next tile into the other buffer, then wait for this tile:
        // each wave has 2 async loads in flight per tile, so asynccnt<=2
        // means tile k's loads are complete (in-order per wave).
        if (k + 1 < TPB) {
            prefetch(tile + 1, (k + 1) & 1);
            __builtin_amdgcn_s_wait_asynccnt(2);
        } else {
            __builtin_amdgcn_s_wait_asynccnt(0);
        }
        __syncthreads();   // cross-wave LDS visibility for tile k

        const float4* l4 = reinterpret_cast<const float4*>(&lbuf[k & 1][0]);
        a = l4[2 * tid + 0];
        b = l4[2 * tid + 1];
        __syncthreads();   // all waves done reading before next overwrite
#else
        const float4* x4 = reinterpret_cast<const float4*>(x) + (size_t)tile * (TILE_PTS / 2);
        a = x4[2 * tid + 0];
        b = x4[2 * tid + 1];
#endif

        float h0[PPT], h1[PPT];
        h0[0] = a.x; h1[0] = a.y;
        h0[1] = a.z; h1[1] = a.w;
        h0[2] = b.x; h1[2] = b.y;
        h0[3] = b.z; h1[3] = b.w;

        // 32-layer chain; params are wave-uniform -> s_load, FMA reads SGPRs.
        for (int l = 0; l < NLAYER; ++l) {
            const float w00 = Ws[4 * l + 0];
            const float w01 = Ws[4 * l + 1];
            const float w10 = Ws[4 * l + 2];
            const float w11 = Ws[4 * l + 3];
            const float b0  = bs[2 * l + 0];
            const float b1  = bs[2 * l + 1];
            const float s0  = sc[2 * l + 0];
            const float s1  = sc[2 * l + 1];
            const float t0  = sh[2 * l + 0];
            const float t1  = sh[2 * l + 1];

#pragma unroll
            for (int p = 0; p < PPT; ++p) {
                float u0 = fmaf(h1[p], w01, fmaf(h0[p], w00, b0));
                float u1 = fmaf(h1[p], w11, fmaf(h0[p], w10, b1));
                h0[p] = fmaf(htanh(u0), s0, t0);
                h1[p] = fmaf(htanh(u1), s1, t1);
            }
        }

        const float wo0 = wo[0];
        const float wo1 = wo[1];
        const float bb  = bo[0];

        float4 o;
        o.x = fmaf(h1[0], wo1, fmaf(h0[0], wo0, bb));
        o.y = fmaf(h1[1], wo1, fmaf(h0[1], wo0, bb));
        o.z = fmaf(h1[2], wo1, fmaf(h0[2], wo0, bb));
        o.w = fmaf(h1[3], wo1, fmaf(h0[3], wo0, bb));
        reinterpret_cast<float4*>(out)[(size_t)tile * BLOCK + tid] = o;
    }
}

extern "C" void kernel_launch(void* const* d_in, const int* in_sizes, int n_in,
                              void* d_out, int out_size, void* d_ws, size_t ws_size,
                              hipStream_t stream) {
    const float* x  = (const float*)d_in[0];  // [B,2]
    const float* Ws = (const float*)d_in[1];  // [L,2,2]
    const float* bs = (const float*)d_in[2];  // [L,2]
    const float* sc = (const float*)d_in[3];  // [L,2]
    const float* sh = (const float*)d_in[4];  // [L,2]
    const float* wo = (const float*)d_in[5];  // [1,2]
    const float* bo = (const float*)d_in[6];  // [1]
    float* out = (float*)d_out;               // [B,1]

    // B = 2^23: 8192 tiles of 1024 points; 2048 persistent blocks x 4 tiles.
    fraud_chain_kernel<<<GRID, BLOCK, 0, stream>>>(x, Ws, bs, sc, sh, wo, bo, out);
}